// SecGELU_63711544869214
// MI455X (gfx1250) — compile-verified
//
#include <hip/hip_runtime.h>
#include <stdint.h>

#define GELU_TABLE_SIZE 4096
#define GELU_SCALE 1024.0f

typedef float v4f __attribute__((ext_vector_type(4)));

__device__ __forceinline__ float gelu_one(float xs, const float* __restrict__ tab) {
    float a = __builtin_fabsf(xs);
    int   c = (int)(a * GELU_SCALE);                       // v_cvt_i32_f32 (trunc)
    c = c < (GELU_TABLE_SIZE - 1) ? c : (GELU_TABLE_SIZE - 1);
    float r = fmaxf(xs, 0.0f);                             // relu
    return r - tab[c];                                     // ds_load_b32 gather
}

__device__ __forceinline__ v4f gelu_vec4(v4f xv, const float* __restrict__ tab) {
    v4f r;
#pragma unroll
    for (int j = 0; j < 4; ++j) r[j] = gelu_one(xv[j], tab);
    return r;
}

__global__ __launch_bounds__(256) void SecGELU_63711544869214_kernel(
        const float* __restrict__ x, const float* __restrict__ table,
        float* __restrict__ out, int nvec, int n) {
    __shared__ float tab[GELU_TABLE_SIZE];

    // ---- stage 16 KB LUT into LDS: 1024 float4s, 256 threads, 4 each ------
    {
        const int t = (int)threadIdx.x;
#if defined(__gfx1250__)
        // CDNA5 async global->LDS DMA path (ASYNCcnt-tracked).
        // vdst = per-lane LDS byte offset (low 32 bits of flat LDS pointer,
        // ISA ch10.2: LDS aperture stores the wave-relative offset in [31:0]);
        // vaddr = 64-bit global address pair.
#pragma unroll
        for (int i = 0; i < 4; ++i) {
            const int idx4 = t + i * 256;                  // float4 index
            unsigned long long gaddr =
                (unsigned long long)(uintptr_t)(table + 4 * idx4);
            unsigned lds_off =
                (unsigned)(uintptr_t)(tab + 4 * idx4);
            asm volatile("global_load_async_to_lds_b128 %0, %1, off"
                         :
                         : "v"(lds_off), "v"(gaddr)
                         : "memory");
        }
        asm volatile("s_wait_asynccnt 0" ::: "memory");
#else
#pragma unroll
        for (int i = 0; i < 4; ++i) {
            const int idx4 = t + i * 256;
            ((v4f*)tab)[idx4] = ((const v4f*)table)[idx4];
        }
#endif
    }
    __syncthreads();

    const int gid    = (int)(blockIdx.x * blockDim.x + threadIdx.x);
    const int stride = (int)(gridDim.x * blockDim.x);
    const v4f* __restrict__ x4   = (const v4f*)x;
    v4f* __restrict__       out4 = (v4f*)out;

    int v = gid;
    // ---- main loop: 4 independent NT b128 loads in flight per lane --------
    for (; v + 3 * stride < nvec; v += 4 * stride) {
        v4f x0 = __builtin_nontemporal_load(x4 + v);
        v4f x1 = __builtin_nontemporal_load(x4 + v + stride);
        v4f x2 = __builtin_nontemporal_load(x4 + v + 2 * stride);
        v4f x3 = __builtin_nontemporal_load(x4 + v + 3 * stride);
        __builtin_nontemporal_store(gelu_vec4(x0, tab), out4 + v);
        __builtin_nontemporal_store(gelu_vec4(x1, tab), out4 + v + stride);
        __builtin_nontemporal_store(gelu_vec4(x2, tab), out4 + v + 2 * stride);
        __builtin_nontemporal_store(gelu_vec4(x3, tab), out4 + v + 3 * stride);
    }
    for (; v < nvec; v += stride) {
        v4f xv = __builtin_nontemporal_load(x4 + v);
        __builtin_nontemporal_store(gelu_vec4(xv, tab), out4 + v);
    }
    // ---- scalar tail if n not a multiple of 4 (not the case here) ---------
    for (int i = nvec * 4 + gid; i < n; i += stride) {
        float xs = x[i];
        out[i] = gelu_one(xs, tab);
    }
}

extern "C" void kernel_launch(void* const* d_in, const int* in_sizes, int n_in,
                              void* d_out, int out_size, void* d_ws, size_t ws_size,
                              hipStream_t stream) {
    (void)n_in; (void)out_size; (void)d_ws; (void)ws_size;
    const float* x     = (const float*)d_in[0];
    const float* table = (const float*)d_in[1];
    float*       out   = (float*)d_out;

    const int n    = in_sizes[0];
    const int nvec = n / 4;

    // Grid: ~16 float4s per thread per sweep, capped so each block amortizes
    // its 16 KB LDS staging; 256 threads = 8 wave32 waves per block.
    int blocks = (nvec + 256 * 16 - 1) / (256 * 16);
    if (blocks < 1) blocks = 1;
    if (blocks > 4096) blocks = 4096;

    SecGELU_63711544869214_kernel<<<dim3(blocks), dim3(256), 0, stream>>>(
        x, table, out, nvec, n);
}